// SisterRegressionROIPooling_50500225466539
// MI455X (gfx1250) — compile-verified
//
#include <hip/hip_runtime.h>
#include <cstdint>

// Feature map geometry (fixed by the reference: conv_out is [1, 512, 20, 31]).
#define FM_H 20
#define FM_W 31
#define FM_C 512
#define FM_PIX (FM_H * FM_W)
#define POOL_P 3
// Max staged window pixels per wave: 80 px * 32 ch * 4 B = 10 KB LDS -> the
// 32-WG/WGP occupancy cap stays reachable. Input ROIs at 1/16 scale span at
// most ~9x7 = 63 px, so the direct-global fallback never triggers here.
#define MAXPIX 80

// ---------------------------------------------------------------------------
// Kernel 1: [C, H*W] -> [H*W, C] transpose so channel is the fast axis.
// 1.27 MB total; negligible cost, buys coalesced lanes in kernel 2.
// ---------------------------------------------------------------------------
__global__ void transpose_chw_to_pxc(const float* __restrict__ in,
                                     float* __restrict__ out) {
  int idx = blockIdx.x * blockDim.x + threadIdx.x;
  if (idx >= FM_PIX * FM_C) return;
  int c = idx & (FM_C - 1);   // idx % 512
  int p = idx >> 9;           // idx / 512
  out[idx] = in[c * FM_PIX + p];
}

// Shared 9-bin quantized max-pool + sum; the element loader is a template so
// the LDS path and the global path stay in *separate* loop nests (pure
// addrspace(3) reads in the LDS version -> ds_load_b32).
template <typename LoadFn>
__device__ __forceinline__ float bins_sum(const int* __restrict__ h0,
                                          const int* __restrict__ h1,
                                          const int* __restrict__ w0,
                                          const int* __restrict__ w1,
                                          LoadFn ld) {
  float sum = 0.0f;
#pragma unroll
  for (int ph = 0; ph < POOL_P; ++ph) {
#pragma unroll
    for (int pw = 0; pw < POOL_P; ++pw) {
      if (h1[ph] > h0[ph] && w1[pw] > w0[pw]) {   // empty bin -> contributes 0
        float mx = -__builtin_huge_valf();
        for (int r = h0[ph]; r < h1[ph]; ++r)
          for (int c = w0[pw]; c < w1[pw]; ++c)
            mx = fmaxf(mx, ld(r, c));
        sum += mx;
      }
    }
  }
  return sum;
}

// ---------------------------------------------------------------------------
// Kernel 2: one wave32 workgroup per (roi, 32-channel group).
//   lane = channel within group (32 lanes -> one contiguous 128 B line).
//   Stage the ROI window into LDS with gfx1250 async global->LDS loads
//   (ASYNCcnt path), wait on s_wait_asynccnt, then run the 9 bins out of LDS.
// ---------------------------------------------------------------------------
__global__ __launch_bounds__(32) void roi_maxpool_avg(
    const float* __restrict__ feat, const float* __restrict__ rois,
    float* __restrict__ out, int nroi, int pix_stride, int ch_stride) {
  __shared__ float smem[MAXPIX * 32];

  const int lane = threadIdx.x;          // 0..31 (wave32)
  const int roi  = blockIdx.x >> 4;
  const int cg   = blockIdx.x & 15;
  if (roi >= nroi) return;
  const int ch = cg * 32 + lane;

  // --- per-ROI scalars (uniform across the wave; batch index is always 0) ---
  const float S = 1.0f / 16.0f;
  const float x1 = rintf(rois[roi * 5 + 1] * S);   // RNE == jnp.round
  const float y1 = rintf(rois[roi * 5 + 2] * S);
  const float x2 = rintf(rois[roi * 5 + 3] * S);
  const float y2 = rintf(rois[roi * 5 + 4] * S);
  const float roi_w = fmaxf(x2 - x1 + 1.0f, 1.0f);
  const float roi_h = fmaxf(y2 - y1 + 1.0f, 1.0f);
  const float bin_w = roi_w / (float)POOL_P;
  const float bin_h = roi_h / (float)POOL_P;

  int h0[POOL_P], h1[POOL_P], w0[POOL_P], w1[POOL_P];
#pragma unroll
  for (int p = 0; p < POOL_P; ++p) {
    h0[p] = (int)fminf(fmaxf(floorf((float)p * bin_h) + y1, 0.0f), (float)FM_H);
    h1[p] = (int)fminf(fmaxf(ceilf((float)(p + 1) * bin_h) + y1, 0.0f), (float)FM_H);
    w0[p] = (int)fminf(fmaxf(floorf((float)p * bin_w) + x1, 0.0f), (float)FM_W);
    w1[p] = (int)fminf(fmaxf(ceilf((float)(p + 1) * bin_w) + x1, 0.0f), (float)FM_W);
  }
  // Bin edges are monotone in p, so the union window is:
  const int r0 = h0[0], r1 = h1[POOL_P - 1];
  const int c0 = w0[0], c1 = w1[POOL_P - 1];
  const int nr = r1 - r0, nc = c1 - c0;
  const int npix = (nr > 0 && nc > 0) ? nr * nc : 0;
  const bool use_lds = (npix > 0) && (npix <= MAXPIX);

  float sum;
  if (use_lds) {
    // --- async-stage window slice: global [pix][32ch] -> LDS [pix][32ch] ---
    const unsigned lds_base = (unsigned)(uintptr_t)(&smem[0]);
    int p = 0;
    for (int r = r0; r < r1; ++r) {
      for (int col = c0; col < c1; ++col) {
        const unsigned goff =
            (unsigned)(((r * FM_W + col) * pix_stride + ch * ch_stride) * 4);
        const unsigned loff = lds_base + (unsigned)((p * 32 + lane) * 4);
        // gfx1250 async global->LDS copy; tracked by ASYNCcnt.
        // GVS addressing: mem = SADDR(64b) + VADDR(32b offset); LDS = VDST.
        asm volatile("global_load_async_to_lds_b32 %0, %1, %2"
                     :: "v"(loff), "v"(goff), "s"(feat)
                     : "memory");
        ++p;
      }
    }
    // Single wave owns this LDS slice -> counter wait suffices (no barrier).
    asm volatile("s_wait_asynccnt 0" ::: "memory");

    // Bins read ONLY from LDS in this branch -> ds_load_b32.
    sum = bins_sum(h0, h1, w0, w1, [&](int r, int c) -> float {
      return smem[((r - r0) * nc + (c - c0)) * 32 + lane];
    });
  } else {
    // Oversized (or empty) window: read straight from L2-resident global.
    sum = bins_sum(h0, h1, w0, w1, [&](int r, int c) -> float {
      return feat[(r * FM_W + c) * pix_stride + ch * ch_stride];
    });
  }

  // mean over the 3x3 grid; reshape(n,-1,4) is a no-op on the flat [n,C] layout
  out[roi * FM_C + ch] = sum / 9.0f;
}

// ---------------------------------------------------------------------------
extern "C" void kernel_launch(void* const* d_in, const int* in_sizes, int n_in,
                              void* d_out, int out_size, void* d_ws, size_t ws_size,
                              hipStream_t stream) {
  const float* conv = (const float*)d_in[0];   // [1, 512, 20, 31] fp32
  const float* rois = (const float*)d_in[1];   // [n, 5] fp32
  float* out = (float*)d_out;                  // [n, 128, 4] fp32 (flat n*512)
  const int nroi = in_sizes[1] / 5;

  const size_t trans_bytes = (size_t)FM_PIX * FM_C * sizeof(float);
  const bool do_transpose = (ws_size >= trans_bytes);

  const float* feat = conv;
  int pix_stride = 1, ch_stride = FM_PIX;      // fallback: native [C, H*W]
  if (do_transpose) {
    float* trans = (float*)d_ws;
    const int tot = FM_PIX * FM_C;
    transpose_chw_to_pxc<<<(tot + 255) / 256, 256, 0, stream>>>(conv, trans);
    feat = trans;
    pix_stride = FM_C;                         // [H*W, C]: lanes coalesced
    ch_stride = 1;
  }

  const int blocks = nroi * (FM_C / 32);       // one wave32 WG per (roi, 32ch)
  roi_maxpool_avg<<<blocks, 32, 0, stream>>>(feat, rois, out, nroi,
                                             pix_stride, ch_stride);
}